// PeakFocusedLoss_16612933501049
// MI455X (gfx1250) — compile-verified
//
#include <hip/hip_runtime.h>
#include <hip/hip_bf16.h>
#include <stdint.h>

// Problem constants (from reference setup_inputs)
#define T_LEN  32768
#define BLOCK  1024          // 32 waves of 32 (wave32)
#define TILE   2048          // floats per staged tile
#define NTILE  (T_LEN / TILE)
#define CAP    512           // max recorded peak regions per array
#define RATIO_F 0.7f

typedef uint32_t u32;
typedef uint64_t u64;
typedef __attribute__((ext_vector_type(4))) u32 v4u;
typedef __attribute__((ext_vector_type(8))) int v8i;
typedef __attribute__((ext_vector_type(4))) int v4i;

#if defined(__has_builtin)
#if __has_builtin(__builtin_amdgcn_tensor_load_to_lds) && __has_builtin(__builtin_amdgcn_s_wait_tensorcnt)
#define HAS_TDM 1
#endif
#endif
#ifndef HAS_TDM
#define HAS_TDM 0
#endif

// ---------------- wave32 / block reductions ----------------
__device__ __forceinline__ float warp_sum(float v) {
  for (int o = 16; o > 0; o >>= 1) v += __shfl_xor(v, o, 32);
  return v;
}
__device__ __forceinline__ float warp_max(float v) {
  for (int o = 16; o > 0; o >>= 1) v = fmaxf(v, __shfl_xor(v, o, 32));
  return v;
}
__device__ float block_sum(float v, float* scratch) {
  v = warp_sum(v);
  int wid = threadIdx.x >> 5, lane = threadIdx.x & 31;
  if (lane == 0) scratch[wid] = v;
  __syncthreads();
  if (wid == 0) {
    float r = scratch[lane];          // 32 waves exactly
    r = warp_sum(r);
    if (lane == 0) scratch[0] = r;
  }
  __syncthreads();
  float r = scratch[0];
  __syncthreads();                    // scratch reusable after return
  return r;
}
__device__ float block_max(float v, float* scratch) {
  v = warp_max(v);
  int wid = threadIdx.x >> 5, lane = threadIdx.x & 31;
  if (lane == 0) scratch[wid] = v;
  __syncthreads();
  if (wid == 0) {
    float r = scratch[lane];
    r = warp_max(r);
    if (lane == 0) scratch[0] = r;
  }
  __syncthreads();
  float r = scratch[0];
  __syncthreads();
  return r;
}

// ---------------- TDM: DMA one contiguous row of TILE f32 into LDS ----------------
#if HAS_TDM
__device__ __forceinline__ void tdm_load_row(const float* gsrc, u32 lds_off) {
  u64 g = (u64)(uintptr_t)gsrc;
  // D# group 0: count=1 | lds_addr | global_addr[56:0] | type=2 at [127:126]
  v4u g0 = { 1u,
             lds_off,
             (u32)g,
             ((u32)(g >> 32) & 0x01FFFFFFu) | 0x80000000u };
  // D# group 1: data_size=2 (4B); tensor_dim0=TILE; tensor_dim1=1;
  //             tile_dim0=TILE; tile_dim1=1; strides=TILE; no pad/iterate/multicast.
  v8i g1 = { 0x00020000,
             (int)((u32)TILE << 16),   // tensor_dim0[15:0] in dw1[31:16]
             0x00010000,               // tensor_dim1[15:0]=1 in dw2[31:16]
             (int)((u32)TILE << 16),   // tile_dim0 in dw3[31:16]
             1,                        // tile_dim1=1, tile_dim2=0
             TILE,                     // tensor_dim0_stride[31:0]
             (int)((u32)TILE << 16),   // tensor_dim1_stride[15:0]
             0 };
  v4i g2 = { 0, 0, 0, 0 };            // dims 2..4 unused (tile dims zero)
  v4i g3 = { 0, 0, 0, 0 };
  v8i g4 = { 0, 0, 0, 0, 0, 0, 0, 0 }; // 6-arg toolchain: extra group, zero-filled
  __builtin_amdgcn_tensor_load_to_lds(g0, g1, g2, g3, g4, 0);
}
#endif

// NaN-masked value of array `a` (0=output,1=target) at index i
__device__ __forceinline__ float sel_val(const float* __restrict__ orow,
                                         const float* __restrict__ trow,
                                         int i, int a) {
  float ov = orow[i], tv = trow[i];
  bool nm = !(ov != ov) && !(tv != tv);
  float v = a ? tv : ov;
  return nm ? v : 0.0f;
}

__global__ __launch_bounds__(BLOCK)
void peak_loss_kernel(const float* __restrict__ outp,
                      const float* __restrict__ tgtp,
                      float* __restrict__ ws) {
  __shared__ float s_o[2][TILE];
  __shared__ float s_t[2][TILE];
  __shared__ float s_red[32];
  __shared__ unsigned s_cnt[2];
  __shared__ float s_pv[2][CAP];
  __shared__ int   s_pi[2][CAP];
  __shared__ float s_sv[2][CAP];
  __shared__ int   s_si[2][CAP];

  const int b = blockIdx.x;
  const float* orow = outp + (size_t)b * T_LEN;
  const float* trow = tgtp + (size_t)b * T_LEN;

  // ---------- Pass 1: max(o), max(t), sum((o-t)^2), TDM-staged ----------
  float mo = -INFINITY, mt = -INFINITY, ss = 0.0f;
#if HAS_TDM
  const bool w0 = (threadIdx.x < 32);   // wave 0 drives the TDM
  if (w0) {
    tdm_load_row(orow, (u32)(uintptr_t)(const void*)&s_o[0][0]);
    tdm_load_row(trow, (u32)(uintptr_t)(const void*)&s_t[0][0]);
  }
  for (int it = 0; it < NTILE; ++it) {
    const int cur = it & 1;
    if (w0) {
      if (it + 1 < NTILE) {
        tdm_load_row(orow + (it + 1) * TILE,
                     (u32)(uintptr_t)(const void*)&s_o[(it + 1) & 1][0]);
        tdm_load_row(trow + (it + 1) * TILE,
                     (u32)(uintptr_t)(const void*)&s_t[(it + 1) & 1][0]);
        __builtin_amdgcn_s_wait_tensorcnt(2);   // tile `it` (2 oldest ops) complete
      } else {
        __builtin_amdgcn_s_wait_tensorcnt(0);
      }
    }
    __syncthreads();                            // publish LDS tile to all waves
#pragma unroll
    for (int e = 0; e < TILE / BLOCK; ++e) {
      int j = threadIdx.x + e * BLOCK;
      float ov = s_o[cur][j], tv = s_t[cur][j];
      bool nm = !(ov != ov) && !(tv != tv);
      ov = nm ? ov : 0.0f; tv = nm ? tv : 0.0f;
      mo = fmaxf(mo, ov); mt = fmaxf(mt, tv);
      float d = ov - tv; ss += d * d;
    }
    __syncthreads();                            // tile consumed; buffer reusable
  }
#else
  for (int i = threadIdx.x; i < T_LEN; i += BLOCK) {
    float ov = orow[i], tv = trow[i];
    bool nm = !(ov != ov) && !(tv != tv);
    ov = nm ? ov : 0.0f; tv = nm ? tv : 0.0f;
    mo = fmaxf(mo, ov); mt = fmaxf(mt, tv);
    float d = ov - tv; ss += d * d;
  }
#endif
  const float max_o = block_max(mo, s_red);
  const float max_t = block_max(mt, s_red);
  const float sumsq = block_sum(ss, s_red);

  // ---------- Pass 2: high/low masked RMSE on target mask (L2-resident reads) ----------
  const bool  has_hi = (max_t > 0.0f);
  const float thr_t  = max_t * RATIO_F;
  float hc = 0.0f, sh = 0.0f, sl = 0.0f;
  if (has_hi) {
    for (int i = threadIdx.x; i < T_LEN; i += BLOCK) {
      float ov = orow[i], tv = trow[i];
      bool nm = !(ov != ov) && !(tv != tv);
      ov = nm ? ov : 0.0f; tv = nm ? tv : 0.0f;
      float d = ov - tv; float sq = d * d;
      if (tv >= thr_t) { hc += 1.0f; sh += sq; } else { sl += sq; }
    }
  }
  hc = block_sum(hc, s_red);
  sh = block_sum(sh, s_red);
  sl = block_sum(sl, s_red);

  const float overall = sqrtf(sumsq / (float)T_LEN);
  float weighted = overall;
  if (has_hi && hc > 0.0f) {
    float hr = sqrtf(sh / hc);
    float lc = (float)T_LEN - hc;
    float lr = (lc > 0.0f) ? sqrtf(sl / lc) : 0.0f;
    weighted = 0.7f * hr + 0.3f * lr;
  }

  // ---------- Pass 3: connected-region peaks for both arrays ----------
  if (threadIdx.x == 0) { s_cnt[0] = 0u; s_cnt[1] = 0u; }
  __syncthreads();
  for (int a = 0; a < 2; ++a) {
    const float mx = a ? max_t : max_o;
    if (mx > 0.0f) {
      const float thr = mx * RATIO_F;
      for (int i = threadIdx.x; i < T_LEN; i += BLOCK) {
        float v = sel_val(orow, trow, i, a);
        if (v >= thr) {
          bool start = (i == 0) || (sel_val(orow, trow, i - 1, a) < thr);
          if (start) {            // walk the run: strict '>' keeps FIRST argmax
            float rmax = v; int rarg = i;
            for (int j = i + 1; j < T_LEN; ++j) {
              float vj = sel_val(orow, trow, j, a);
              if (vj < thr) break;
              if (vj > rmax) { rmax = vj; rarg = j; }
            }
            unsigned pos = atomicAdd(&s_cnt[a], 1u);
            if (pos < CAP) { s_pv[a][pos] = rmax; s_pi[a][pos] = rarg; }
          }
        }
      }
    }
  }
  __syncthreads();

  // ---------- Stable rank sort: value desc, index asc (order-independent) ----------
  unsigned nR0 = s_cnt[0]; if (nR0 > CAP) nR0 = CAP;
  unsigned nR1 = s_cnt[1]; if (nR1 > CAP) nR1 = CAP;
  for (int a = 0; a < 2; ++a) {
    const unsigned n = a ? nR1 : nR0;
    for (unsigned k = threadIdx.x; k < n; k += BLOCK) {
      float vk = s_pv[a][k]; int ik = s_pi[a][k];
      unsigned rank = 0;
      for (unsigned j = 0; j < n; ++j) {
        float vj = s_pv[a][j]; int ij = s_pi[a][j];
        if (vj > vk || (vj == vk && ij < ik)) rank++;
      }
      s_sv[a][rank] = vk; s_si[a][rank] = ik;
    }
  }
  __syncthreads();

  // ---------- Pair top-m peaks, finalize ----------
  if (threadIdx.x == 0) {
    unsigned m = (nR0 < nR1) ? nR0 : nR1;
    float pe = 0.0f, te = 0.0f;
    for (unsigned k = 0; k < m; ++k) {
      pe += fabsf(s_sv[0][k] - s_sv[1][k]);
      te += fabsf((float)(s_si[0][k] - s_si[1][k]));
    }
    float denom = (float)(m > 0 ? m : 1u);
    pe /= denom; te /= denom;
    ws[b] = 0.5f * overall + 0.5f * weighted + 2.0f * pe + 1.0f * te;
  }
}

__global__ void mean_kernel(const float* __restrict__ ws, float* __restrict__ out, int n) {
  if (threadIdx.x == 0 && blockIdx.x == 0) {
    float s = 0.0f;
    for (int i = 0; i < n; ++i) s += ws[i];
    out[0] = s / (float)n;
  }
}

extern "C" void kernel_launch(void* const* d_in, const int* in_sizes, int n_in,
                              void* d_out, int out_size, void* d_ws, size_t ws_size,
                              hipStream_t stream) {
  const float* outp = (const float*)d_in[0];
  const float* tgtp = (const float*)d_in[1];
  const int B = in_sizes[0] / T_LEN;
  float* ws = (float*)d_ws;
  peak_loss_kernel<<<B, BLOCK, 0, stream>>>(outp, tgtp, ws);
  mean_kernel<<<1, 32, 0, stream>>>(ws, (float*)d_out, B);
}